// GAT_12266426597844
// MI455X (gfx1250) — compile-verified
//
#include <hip/hip_runtime.h>
#include <math.h>

#define N_NODES 50000
#define IN_F    128
#define NHEAD   8
#define HDIM    32
#define HID     256
#define NEG_SLOPE 0.2f

typedef __attribute__((ext_vector_type(2))) float v2f;
typedef __attribute__((ext_vector_type(8))) float v8f;

// ---------------------------------------------------------------------------
// GEMM: Out[M,Nn] = X[M,K] @ W[K,Nn]   via V_WMMA_F32_16X16X4_F32 (wave32)
// M % 16 == 0, Nn % 16 == 0, K % 4 == 0, K <= 256 (power of two here).
// Block = 256 threads = 8 waves. Each block owns one 16-row strip of X
// (staged in LDS, shared by all 8 waves); each wave sweeps 16-col tiles.
// Per-lane WMMA layouts (ISA 7.12.2):
//   A 16x4 : lane = M (0..15) in each half; v0 = A[M][k0+2*half], v1 = +1
//   B 4x16 : lane = N (0..15) in each half; v0 = B[k0+2*half][N], v1 = +1
//   C 16x16: vgpr j = row (j + 8*half), col = lane&15
// ---------------------------------------------------------------------------
__global__ __launch_bounds__(256) void gemm_wmma_f32(
    const float* __restrict__ X, const float* __restrict__ W,
    float* __restrict__ Out, int M, int K, int Nn)
{
  __shared__ float As[16 * 256];              // 16 rows x K (K <= 256)
  const int m0    = blockIdx.x << 4;
  const int tid   = threadIdx.x;
  const int kbits = 31 - __clz(K);            // K is a power of two (128/256)

  // cooperative load of the 16 x K A-tile into LDS (fully coalesced rows)
  for (int idx = tid; idx < (16 << kbits); idx += 256) {
    const int r = idx >> kbits;
    const int c = idx & (K - 1);
    As[(r << kbits) + c] = X[(size_t)(m0 + r) * K + c];
  }
  __syncthreads();

  const int wave = tid >> 5;
  const int lane = tid & 31;
  const int half = lane >> 4;                 // K-half selector for A/B
  const int r    = lane & 15;                 // M index for A, N index for B

  for (int ntile = wave; ntile < (Nn >> 4); ntile += 8) {
    const int n0 = ntile << 4;
    v8f acc = {};
    for (int k0 = 0; k0 < K; k0 += 4) {
      const int ka = k0 + (half << 1);
      v2f a, b;
      a.x = As[(r << kbits) + ka];
      a.y = As[(r << kbits) + ka + 1];
      b.x = W[(size_t)ka * Nn + n0 + r];
      b.y = W[(size_t)(ka + 1) * Nn + n0 + r];
      acc = __builtin_amdgcn_wmma_f32_16x16x4_f32(
          /*neg_a=*/false, a, /*neg_b=*/false, b,
          /*c_mod=*/(short)0, acc, /*reuse_a=*/false, /*reuse_b=*/false);
    }
#pragma unroll
    for (int j = 0; j < 8; ++j) {
      const int row = m0 + j + (half << 3);
      Out[(size_t)row * Nn + n0 + r] = acc[j];
    }
  }
}

// ---------------------------------------------------------------------------
// Per-(node, head) attention logits: el = <h, a_l>, er = <h, a_r>
// ---------------------------------------------------------------------------
__global__ __launch_bounds__(256) void attn_scores(
    const float* __restrict__ Hf,
    const float* __restrict__ a_l, const float* __restrict__ a_r,
    float* __restrict__ el, float* __restrict__ er, int total)
{
  const int i = blockIdx.x * 256 + threadIdx.x;   // i = n*NHEAD + h
  if (i >= total) return;
  const int n = i >> 3;
  const int h = i & 7;
  const float* __restrict__ hp = Hf + (size_t)n * HID + h * HDIM;
  const float* __restrict__ wl = a_l + h * HDIM;
  const float* __restrict__ wr = a_r + h * HDIM;
  float sl = 0.f, sr = 0.f;
#pragma unroll
  for (int d = 0; d < HDIM; ++d) {
    const float v = hp[d];
    sl += v * wl[d];
    sr += v * wr[d];
  }
  el[i] = sl;
  er[i] = sr;
}

__global__ __launch_bounds__(256) void fill_f32(float* __restrict__ p, float v, int count)
{
  const int i = blockIdx.x * 256 + threadIdx.x;
  if (i < count) p[i] = v;
}

__device__ __forceinline__ float leaky(float x)
{
  return x > 0.f ? x : NEG_SLOPE * x;
}

// Atomic float max via the signed/unsigned int ordering trick (init = -inf)
__device__ __forceinline__ void atomicMaxF(float* addr, float val)
{
  if (val >= 0.f) atomicMax((int*)addr, __float_as_int(val));
  else            atomicMin((unsigned int*)addr, (unsigned int)__float_as_int(val));
}

// ---------------------------------------------------------------------------
// Pass 1: segment-max of leaky_relu(el[src]+er[dst]) over incoming edges
// ---------------------------------------------------------------------------
__global__ __launch_bounds__(256) void edge_max(
    const int* __restrict__ src, const int* __restrict__ dst,
    const float* __restrict__ el, const float* __restrict__ er,
    float* __restrict__ m, int total)     // total = E * NHEAD
{
  const int i = blockIdx.x * 256 + threadIdx.x;
  if (i >= total) return;
  const int e = i >> 3;
  const int h = i & 7;
  const int s = src[e], d = dst[e];
  const float x = leaky(el[s * NHEAD + h] + er[d * NHEAD + h]);
  atomicMaxF(&m[d * NHEAD + h], x);
}

// ---------------------------------------------------------------------------
// Pass 2: ex = exp(e - m[dst]); stash per-edge ex; segment-sum into s[dst]
// ---------------------------------------------------------------------------
__global__ __launch_bounds__(256) void edge_expsum(
    const int* __restrict__ src, const int* __restrict__ dst,
    const float* __restrict__ el, const float* __restrict__ er,
    const float* __restrict__ m, float* __restrict__ exb,
    float* __restrict__ ssum, int total)  // total = E * NHEAD
{
  const int i = blockIdx.x * 256 + threadIdx.x;
  if (i >= total) return;
  const int e = i >> 3;
  const int h = i & 7;
  const int s = src[e], d = dst[e];
  const float x  = leaky(el[s * NHEAD + h] + er[d * NHEAD + h]);
  const float ex = __expf(x - m[d * NHEAD + h]);
  exb[i] = ex;
  atomicAdd(&ssum[d * NHEAD + h], ex);
}

// ---------------------------------------------------------------------------
// Pass 3: out[dst] += (ex/s[dst]) * h[src].  One wave32 per edge; lane = dim
// within head (HDIM == warpSize == 32) -> coalesced 128B loads + atomics.
// ---------------------------------------------------------------------------
__global__ __launch_bounds__(256) void edge_aggregate(
    const int* __restrict__ src, const int* __restrict__ dst,
    const float* __restrict__ Hf, const float* __restrict__ exb,
    const float* __restrict__ ssum, float* __restrict__ agg, int E)
{
  const int e = blockIdx.x * 8 + (threadIdx.x >> 5);
  if (e >= E) return;
  const int lane = threadIdx.x & 31;
  const int s = src[e], d = dst[e];
  const float* __restrict__ hs   = Hf  + (size_t)s * HID;
  float* __restrict__       outp = agg + (size_t)d * HID;
#pragma unroll
  for (int h = 0; h < NHEAD; ++h) {
    const float alpha = exb[(size_t)e * NHEAD + h] / ssum[d * NHEAD + h];
    atomicAdd(&outp[h * HDIM + lane], alpha * hs[h * HDIM + lane]);
  }
}

// ---------------------------------------------------------------------------
// x[i] = (x[i] + b[i % 256]) , optionally ELU (alpha = 1)
// ---------------------------------------------------------------------------
__global__ __launch_bounds__(256) void bias_act(
    float* __restrict__ x, const float* __restrict__ b, int total, int do_elu)
{
  const int i = blockIdx.x * 256 + threadIdx.x;
  if (i >= total) return;
  float v = x[i] + b[i & (HID - 1)];
  if (do_elu) v = v > 0.f ? v : (__expf(v) - 1.0f);
  x[i] = v;
}

// ---------------------------------------------------------------------------
extern "C" void kernel_launch(void* const* d_in, const int* in_sizes, int n_in,
                              void* d_out, int out_size, void* d_ws, size_t ws_size,
                              hipStream_t stream)
{
  const float* in_feat = (const float*)d_in[0];
  const int*   src     = (const int*)  d_in[1];
  const int*   dst     = (const int*)  d_in[2];
  const float* W1      = (const float*)d_in[3];
  const float* a_l1    = (const float*)d_in[4];
  const float* a_r1    = (const float*)d_in[5];
  const float* b1      = (const float*)d_in[6];
  const float* W2      = (const float*)d_in[7];
  const float* a_l2    = (const float*)d_in[8];
  const float* a_r2    = (const float*)d_in[9];
  const float* b2      = (const float*)d_in[10];

  const int E  = in_sizes[1];          // 850000 (800k random + 50k self loops)
  const int NH = N_NODES * NHEAD;      // 400000
  const int NF = N_NODES * HID;        // 12.8M
  const int EH = E * NHEAD;

  // workspace carve-up (floats)
  float* ws   = (float*)d_ws;
  float* hbuf = ws;                        // N*HID   (h1, reused as h2)
  float* agg  = hbuf + (size_t)NF;         // N*HID   (layer-1 aggregate -> x2)
  float* el   = agg  + (size_t)NF;         // N*NHEAD
  float* er   = el   + (size_t)NH;         // N*NHEAD
  float* mmax = er   + (size_t)NH;         // N*NHEAD
  float* ssum = mmax + (size_t)NH;         // N*NHEAD
  float* exb  = ssum + (size_t)NH;         // E*NHEAD

  const int gemm_blocks = N_NODES / 16;        // 3125
  const int nh_blocks   = (NH + 255) / 256;
  const int nf_blocks   = (NF + 255) / 256;
  const int eh_blocks   = (EH + 255) / 256;
  const int agg_blocks  = (E + 7) / 8;

  float* out = (float*)d_out;

  // ================= layer 1 =================
  gemm_wmma_f32<<<gemm_blocks, 256, 0, stream>>>(in_feat, W1, hbuf, N_NODES, IN_F, HID);
  attn_scores  <<<nh_blocks, 256, 0, stream>>>(hbuf, a_l1, a_r1, el, er, NH);

  fill_f32<<<nh_blocks, 256, 0, stream>>>(mmax, -INFINITY, NH);
  fill_f32<<<nh_blocks, 256, 0, stream>>>(ssum, 0.0f, NH);
  fill_f32<<<nf_blocks, 256, 0, stream>>>(agg, 0.0f, NF);

  edge_max    <<<eh_blocks, 256, 0, stream>>>(src, dst, el, er, mmax, EH);
  edge_expsum <<<eh_blocks, 256, 0, stream>>>(src, dst, el, er, mmax, exb, ssum, EH);
  edge_aggregate<<<agg_blocks, 256, 0, stream>>>(src, dst, hbuf, exb, ssum, agg, E);

  bias_act<<<nf_blocks, 256, 0, stream>>>(agg, b1, NF, /*elu=*/1);   // agg -> x2

  // ================= layer 2 =================
  gemm_wmma_f32<<<gemm_blocks, 256, 0, stream>>>(agg, W2, hbuf, N_NODES, HID, HID);
  attn_scores  <<<nh_blocks, 256, 0, stream>>>(hbuf, a_l2, a_r2, el, er, NH);

  fill_f32<<<nh_blocks, 256, 0, stream>>>(mmax, -INFINITY, NH);
  fill_f32<<<nh_blocks, 256, 0, stream>>>(ssum, 0.0f, NH);
  fill_f32<<<nf_blocks, 256, 0, stream>>>(out, 0.0f, NF);

  edge_max    <<<eh_blocks, 256, 0, stream>>>(src, dst, el, er, mmax, EH);
  edge_expsum <<<eh_blocks, 256, 0, stream>>>(src, dst, el, er, mmax, exb, ssum, EH);
  edge_aggregate<<<agg_blocks, 256, 0, stream>>>(src, dst, hbuf, exb, ssum, out, E);

  bias_act<<<nf_blocks, 256, 0, stream>>>(out, b2, NF, /*elu=*/0);
}